// GCN_566935683470
// MI455X (gfx1250) — compile-verified
//
#include <hip/hip_runtime.h>

#define N_NODES 50000
#define FDIM    128      // IN_DIM == HID == 128 (all SpMM feature dims)
#define OUT_DIM 64

typedef __attribute__((ext_vector_type(2))) float        v2f;
typedef __attribute__((ext_vector_type(4))) float        v4f;
typedef __attribute__((ext_vector_type(8))) float        v8f;
typedef __attribute__((ext_vector_type(4))) unsigned int v4u;
typedef __attribute__((ext_vector_type(4))) int          v4i;
typedef __attribute__((ext_vector_type(8))) int          v8i;

#if defined(__HIP_DEVICE_COMPILE__) && defined(__has_builtin)
#if __has_builtin(__builtin_amdgcn_tensor_load_to_lds) && \
    __has_builtin(__builtin_amdgcn_s_wait_tensorcnt)
#define HAVE_TDM 1
#endif
#endif

// ---------------------------------------------------------------------------
// Zero-fill (float4 stores)
// ---------------------------------------------------------------------------
__global__ void zero_kernel(float* __restrict__ p, int n4) {
    int i = blockIdx.x * blockDim.x + threadIdx.x;
    if (i < n4) {
        v4f z = {0.f, 0.f, 0.f, 0.f};
        ((v4f*)p)[i] = z;
    }
}

// ---------------------------------------------------------------------------
// COO SpMM: out[row] += w * H[col]   (one wave32 per edge, float4 per lane)
// ---------------------------------------------------------------------------
__global__ void spmm_kernel(const int* __restrict__ erow,
                            const int* __restrict__ ecol,
                            const float* __restrict__ ew,
                            const float* __restrict__ H,
                            float* __restrict__ out,
                            int E) {
    int wid  = (blockIdx.x * blockDim.x + threadIdx.x) >> 5;
    int lane = threadIdx.x & 31;
    if (wid >= E) return;

    int   r = erow[wid];
    int   c = ecol[wid];
    float w = ew[wid];

    v4f v = ((const v4f*)(H + (long)c * FDIM))[lane];
    float* dst = out + (long)r * FDIM + lane * 4;

    __hip_atomic_fetch_add(dst + 0, v.x * w, __ATOMIC_RELAXED, __HIP_MEMORY_SCOPE_AGENT);
    __hip_atomic_fetch_add(dst + 1, v.y * w, __ATOMIC_RELAXED, __HIP_MEMORY_SCOPE_AGENT);
    __hip_atomic_fetch_add(dst + 2, v.z * w, __ATOMIC_RELAXED, __HIP_MEMORY_SCOPE_AGENT);
    __hip_atomic_fetch_add(dst + 3, v.w * w, __ATOMIC_RELAXED, __HIP_MEMORY_SCOPE_AGENT);
}

// ---------------------------------------------------------------------------
// TDM: 2D tile load Global -> LDS (rows x cols fp32, row stride strideElems).
// D# layout per CDNA5 ISA:
//   group0: [1:0]=count=1 | [63:32]=lds_addr | [120:64]=global_addr | [127:126]=type=2
//   group1: [17:16]=data_size(2 -> 4B) | [79:48]=tensor_dim0 | [111:80]=tensor_dim1
//           [127:112]=tile_dim0 | [143:128]=tile_dim1 | [207:160]=tensor_dim0_stride
//   groups 2/3: zero (2D tensor)
// ---------------------------------------------------------------------------
#if defined(HAVE_TDM)
__device__ __forceinline__ void tdm_load_2d(const void* gsrc, unsigned ldsAddr,
                                            unsigned rows, unsigned cols,
                                            unsigned strideElems) {
    unsigned long long ga = (unsigned long long)(uintptr_t)gsrc;
    v4u g0 = { 1u,                                    // count=1, is_restore=0
               ldsAddr,                               // lds_addr (bytes)
               (unsigned)ga,                          // global_addr[31:0]
               (unsigned)(ga >> 32) | (2u << 30) };   // global_addr[56:32] | type=2
    v8i g1 = { (int)(2u << 16),                       // data_size = 2 (4 bytes)
               (int)((cols & 0xFFFFu) << 16),         // tensor_dim0[15:0]
               (int)((rows & 0xFFFFu) << 16),         // tensor_dim1[15:0]
               (int)((cols & 0xFFFFu) << 16),         // tile_dim0
               (int)(rows & 0xFFFFu),                 // tile_dim1 (tile_dim2=0)
               (int)strideElems,                      // tensor_dim0_stride[31:0]
               0, 0 };
    v4i gz = {0, 0, 0, 0};
#if __clang_major__ >= 23
    v8i gz8 = {0, 0, 0, 0, 0, 0, 0, 0};
    __builtin_amdgcn_tensor_load_to_lds(g0, g1, gz, gz, gz8, 0);
#else
    __builtin_amdgcn_tensor_load_to_lds(g0, g1, gz, gz, 0);
#endif
}
#endif

// ---------------------------------------------------------------------------
// Dense GEMM + bias (+ReLU) with V_WMMA_F32_16X16X4_F32.
//   One wave per 16x16 output tile; NCOLS/16 waves per block; W staged in LDS.
// Fragment layouts per CDNA5 ISA 7.12.2:
//   A 16x4 f32 : lanes 0-15 M=lane, {K0,K1}; lanes 16-31 M=lane-16, {K2,K3}
//   B 4x16 f32 : lanes 0-15 N=lane, {K0,K1}; lanes 16-31 N=lane-16, {K2,K3}
//   C/D 16x16  : VGPR r -> M = r + 8*(lane>=16), N = lane&15
// ---------------------------------------------------------------------------
template <int KDIM, int NCOLS, bool RELU>
__global__ void wmma_gemm_kernel(const float* __restrict__ A,
                                 const float* __restrict__ W,
                                 const float* __restrict__ bias,
                                 float* __restrict__ out) {
    __shared__ float sW[KDIM * NCOLS];

    // ---- stage W into LDS -------------------------------------------------
#if defined(HAVE_TDM)
    if (threadIdx.x < 32) {                       // wave 0 drives the TDM
        tdm_load_2d(W, (unsigned)(uintptr_t)&sW[0], KDIM, NCOLS, NCOLS);
        __builtin_amdgcn_s_wait_tensorcnt(0);
    }
#else
    for (int i = threadIdx.x; i < (KDIM * NCOLS) / 4; i += blockDim.x)
        ((v4f*)sW)[i] = ((const v4f*)W)[i];
#endif
    __syncthreads();

    // ---- WMMA main loop ---------------------------------------------------
    const int waveId  = threadIdx.x >> 5;          // NCOLS/16 waves per block
    const int lane    = threadIdx.x & 31;
    const int half    = lane >> 4;                 // 0: lanes 0-15, 1: lanes 16-31
    const int l       = lane & 15;
    const int rowBase = blockIdx.x * 16;
    const int colBase = waveId * 16;

    const float* Arow = A + (long)(rowBase + l) * KDIM;
    const float* Bcol = sW + colBase + l;

    v8f acc = {};
#pragma unroll
    for (int k = 0; k < KDIM; k += 4) {
        v2f a = *(const v2f*)(Arow + k + 2 * half);   // {K+2h, K+2h+1}
        v2f b;
        b.x = Bcol[(k + 2 * half)     * NCOLS];
        b.y = Bcol[(k + 2 * half + 1) * NCOLS];
        acc = __builtin_amdgcn_wmma_f32_16x16x4_f32(
            /*neg_a=*/false, a, /*neg_b=*/false, b,
            /*c_mod=*/(short)0, acc, /*reuse_a=*/false, /*reuse_b=*/false);
    }

    const float bv = bias[colBase + l];
#pragma unroll
    for (int r = 0; r < 8; ++r) {
        float v = acc[r] + bv;
        if (RELU) v = fmaxf(v, 0.f);
        out[(long)(rowBase + r + 8 * half) * NCOLS + colBase + l] = v;
    }
}

// ---------------------------------------------------------------------------
// Host-side launcher
// ---------------------------------------------------------------------------
extern "C" void kernel_launch(void* const* d_in, const int* in_sizes, int n_in,
                              void* d_out, int out_size, void* d_ws, size_t ws_size,
                              hipStream_t stream) {
    const float* X    = (const float*)d_in[0];
    const int*   erow = (const int*)d_in[1];
    const int*   ecol = (const int*)d_in[2];
    const float* ew   = (const float*)d_in[3];
    const float* W1   = (const float*)d_in[4];
    const float* b1   = (const float*)d_in[5];
    const float* W2   = (const float*)d_in[6];
    const float* b2   = (const float*)d_in[7];
    const float* W3   = (const float*)d_in[8];
    const float* b3   = (const float*)d_in[9];
    float* out = (float*)d_out;

    const int E = in_sizes[1];

    float* bufA = (float*)d_ws;                          // SpMM accumulator
    float* bufB = bufA + (size_t)N_NODES * FDIM;         // activation

    const int n4 = N_NODES * FDIM / 4;
    dim3 zgrid((n4 + 255) / 256);
    dim3 sgrid(((size_t)E * 32 + 255) / 256);
    dim3 ggrid(N_NODES / 16);                            // 3125

    // ---- layer 1: SpMM(X) -> GEMM(W1)+b1 -> ReLU
    zero_kernel<<<zgrid, 256, 0, stream>>>(bufA, n4);
    spmm_kernel<<<sgrid, 256, 0, stream>>>(erow, ecol, ew, X, bufA, E);
    wmma_gemm_kernel<FDIM, FDIM, true><<<ggrid, 256, 0, stream>>>(bufA, W1, b1, bufB);

    // ---- layer 2
    zero_kernel<<<zgrid, 256, 0, stream>>>(bufA, n4);
    spmm_kernel<<<sgrid, 256, 0, stream>>>(erow, ecol, ew, bufB, bufA, E);
    wmma_gemm_kernel<FDIM, FDIM, true><<<ggrid, 256, 0, stream>>>(bufA, W2, b2, bufB);

    // ---- layer 3 (no ReLU), writes d_out [N_NODES x OUT_DIM]
    zero_kernel<<<zgrid, 256, 0, stream>>>(bufA, n4);
    spmm_kernel<<<sgrid, 256, 0, stream>>>(erow, ecol, ew, bufB, bufA, E);
    wmma_gemm_kernel<FDIM, OUT_DIM, false><<<ggrid, 128, 0, stream>>>(bufA, W3, b3, out);
}